// ModifySupConLoss_54150947668152
// MI455X (gfx1250) — compile-verified
//
#include <hip/hip_runtime.h>
#include <hip/hip_bf16.h>
#include <math.h>

// ---------------------------------------------------------------------------
// ModifySupConLoss for MI455X (gfx1250, wave32)
//
// loss = [ 8*sum_r log(denom_r) - 4*sum_r sa_r/T - sum_k min(s_k,1)/T ] / 65536
//   denom_r = rowsum_r(exp(S/T)) - exp(selfdot_r/T) + 1e-6,  S = Zn Zn^T (fp32)
// Heavy part: 8192x8192x128 fp32 Gram matrix via V_WMMA_F32_16X16X4_F32.
// ---------------------------------------------------------------------------

typedef __attribute__((ext_vector_type(2))) float v2f;
typedef __attribute__((ext_vector_type(4))) float v4f;
typedef __attribute__((ext_vector_type(8))) float v8f;

#define BATCH   4096
#define TWOB    8192
#define DIM     128
#define NNB     16384      // B * NEARBY rows of zn
#define NPAIR   32768      // 2B * NEARBY positive (aug,nearby) pairs
#define NTILES  512        // TWOB / 16
#define INV_TEMP (1.0f / 0.07f)
#define EPS_F    1e-6f
#define COS_EPS_F 1e-8f

__device__ __forceinline__ float wave_sum32(float v) {
    v += __shfl_xor(v, 1);
    v += __shfl_xor(v, 2);
    v += __shfl_xor(v, 4);
    v += __shfl_xor(v, 8);
    v += __shfl_xor(v, 16);
    return v;
}

// butterfly within each 16-lane half (never crosses lane bit 4)
__device__ __forceinline__ float half_sum16(float v) {
    v += __shfl_xor(v, 1);
    v += __shfl_xor(v, 2);
    v += __shfl_xor(v, 4);
    v += __shfl_xor(v, 8);
    return v;
}

__device__ __forceinline__ float dot4(v4f a, v4f b) {
    return fmaf(a.x, b.x, fmaf(a.y, b.y, fmaf(a.z, b.z, a.w * b.w)));
}

// ---------------------------------------------------------------------------
// 1) normalize za = concat(z1, z2) -> zan  (one wave per row)
// ---------------------------------------------------------------------------
__launch_bounds__(256)
__global__ void norm_za_kernel(const float* __restrict__ z1,
                               const float* __restrict__ z2,
                               float* __restrict__ zan) {
    const int lane = threadIdx.x & 31;
    const int wv   = threadIdx.x >> 5;
    const int row  = blockIdx.x * 8 + wv;                 // 0..8191
    const float* src = (row < BATCH) ? (z1 + (size_t)row * DIM)
                                     : (z2 + (size_t)(row - BATCH) * DIM);
    v4f x = *(const v4f*)(src + lane * 4);
    float ss = wave_sum32(dot4(x, x));
    float inv = 1.0f / fmaxf(sqrtf(ss), COS_EPS_F);
    x.x *= inv; x.y *= inv; x.z *= inv; x.w *= inv;
    *(v4f*)(zan + (size_t)row * DIM + lane * 4) = x;
}

// ---------------------------------------------------------------------------
// 2) per-row 1/max(||zn||, eps)
// ---------------------------------------------------------------------------
__launch_bounds__(256)
__global__ void zn_invnorm_kernel(const float* __restrict__ zn,
                                  float* __restrict__ invn) {
    const int lane = threadIdx.x & 31;
    const int wv   = threadIdx.x >> 5;
    const int row  = blockIdx.x * 8 + wv;                 // 0..16383
    v4f x = *(const v4f*)(zn + (size_t)row * DIM + lane * 4);
    float ss = wave_sum32(dot4(x, x));
    if (lane == 0) invn[row] = 1.0f / fmaxf(sqrtf(ss), COS_EPS_F);
}

// ---------------------------------------------------------------------------
// 3) Gram row-sum-of-exp via fp32 WMMA (16x16x4, K=128 -> 32 chained WMMAs)
//
//    K-partition remap: op kk covers columns {2kk, 2kk+1, 64+2kk, 65+2kk}.
//    Lanes 0-15 (hf=0) supply the K=0,1 pair (cols 2kk,2kk+1), lanes 16-31
//    (hf=1) supply K=2,3 (cols 64+2kk,65+2kk).  The 32 ops cover all 128
//    columns disjointly, so the chained accumulation equals the full dot.
//    => each lane's operand data is a CONTIGUOUS 256B half-row: 16 b128 loads
//       per tile, batched into registers before the 32 back-to-back WMMAs.
//
//    Loop control is forced scalar (readfirstlane) so the column loop is
//    s_cmp/s_cbranch with EXEC all-1s throughout (WMMA requirement) and no
//    per-iteration exec-mask juggling.
//
//    C/D layout: VGPR v: lanes 0-15 -> M=v, lanes 16-31 -> M=v+8; N=lane&15
// ---------------------------------------------------------------------------
__launch_bounds__(256)
__global__ void gram_rowsum_kernel(const float* __restrict__ zan,
                                   float* __restrict__ rowsum) {
    __shared__ float part[8 * 16];
    const int tid  = threadIdx.x;
    const int lane = tid & 31;
    const int wv   = __builtin_amdgcn_readfirstlane(tid) >> 5;   // scalar wave id
    const int rt   = blockIdx.x;                          // 0..511 row tile
    const int m    = lane & 15;
    const int hf   = lane >> 4;

    // persistent A operand: this lane's contiguous half-row of each A row
    const float* Arow = zan + (size_t)(rt * 16 + m) * DIM + hf * 64;
    v4f A4[16];
#pragma unroll
    for (int i = 0; i < 16; ++i) A4[i] = *(const v4f*)(Arow + 4 * i);

    float rs[8] = {0.f, 0.f, 0.f, 0.f, 0.f, 0.f, 0.f, 0.f};

    for (int ct = wv; ct < NTILES; ct += 8) {
        const float* Brow = zan + (size_t)(ct * 16 + m) * DIM + hf * 64;
        // branchless next-tile cache hint (clamped; last iter re-touches self)
        int ctn = ct + 8 < NTILES ? ct + 8 : ct;
        __builtin_prefetch(zan + (size_t)(ctn * 16 + m) * DIM + hf * 64, 0, 1);

        // batched tile load: 16 x global_load_b128, then WMMA chain
        v4f B4[16];
#pragma unroll
        for (int i = 0; i < 16; ++i) B4[i] = *(const v4f*)(Brow + 4 * i);

        v8f acc = {};
#pragma unroll
        for (int i = 0; i < 16; ++i) {
            v2f a0 = A4[i].xy, b0 = B4[i].xy;
            acc = __builtin_amdgcn_wmma_f32_16x16x4_f32(
                false, a0, false, b0, (short)0, acc, false, false);
            v2f a1 = A4[i].zw, b1 = B4[i].zw;
            acc = __builtin_amdgcn_wmma_f32_16x16x4_f32(
                false, a1, false, b1, (short)0, acc, false, false);
        }
#pragma unroll
        for (int v = 0; v < 8; ++v)
            rs[v] += __expf(acc[v] * INV_TEMP);
    }

    // reduce across the 16 columns held by each 16-lane half
#pragma unroll
    for (int v = 0; v < 8; ++v) {
        float s = half_sum16(rs[v]);
        if (m == 0) part[wv * 16 + hf * 8 + v] = s;       // row_local = v + 8*hf
    }
    __syncthreads();
    if (tid < 16) {
        float t = 0.f;
#pragma unroll
        for (int w = 0; w < 8; ++w) t += part[w * 16 + tid];
        rowsum[rt * 16 + tid] = t;                        // exclusive writer
    }
}

// ---------------------------------------------------------------------------
// 4) per-row term: 8*log(denom) - 4*sim_aug/T   (one wave per row)
// ---------------------------------------------------------------------------
__launch_bounds__(256)
__global__ void terma_kernel(const float* __restrict__ zan,
                             const float* __restrict__ rowsum,
                             float* __restrict__ termA) {
    const int lane = threadIdx.x & 31;
    const int wv   = threadIdx.x >> 5;
    const int row  = blockIdx.x * 8 + wv;                 // 0..8191
    const int pr   = (row + BATCH) & (TWOB - 1);          // (r+B) mod 2B
    v4f x = *(const v4f*)(zan + (size_t)row * DIM + lane * 4);
    v4f y = *(const v4f*)(zan + (size_t)pr  * DIM + lane * 4);
    float sd = wave_sum32(dot4(x, x));                    // self cosine (~1, fp32)
    float sa = wave_sum32(dot4(x, y));                    // aug positive
    if (lane == 0) {
        float denom = rowsum[row] - __expf(sd * INV_TEMP) + EPS_F;
        termA[row] = 8.0f * __logf(denom) - 4.0f * (sa * INV_TEMP);
    }
}

// ---------------------------------------------------------------------------
// 5) per-pair nearby positive: min(cos(za_n[p], zn[q]), 1)/T
//    pair set = diagonals of sim_an at offsets {-B, 0, B, 2B, 3B}
// ---------------------------------------------------------------------------
__launch_bounds__(256)
__global__ void termp_kernel(const float* __restrict__ zan,
                             const float* __restrict__ zn,
                             const float* __restrict__ invn,
                             float* __restrict__ termP) {
    const int lane = threadIdx.x & 31;
    const int wv   = threadIdx.x >> 5;
    const int k    = blockIdx.x * 8 + wv;                 // 0..32767
    int p, q;
    if (k < 4096)        { p = BATCH + k;      q = k; }            // offset -B
    else if (k < 12288)  { int j = k - 4096;  p = j; q = j; }      // offset 0
    else if (k < 20480)  { int j = k - 12288; p = j; q = j + 4096; }  // +B
    else if (k < 28672)  { int j = k - 20480; p = j; q = j + 8192; }  // +2B
    else                 { int j = k - 28672; p = j; q = j + 12288; } // +3B (len B)
    v4f x = *(const v4f*)(zan + (size_t)p * DIM + lane * 4);
    v4f y = *(const v4f*)(zn  + (size_t)q * DIM + lane * 4);
    float d = wave_sum32(dot4(x, y));
    if (lane == 0) {
        float s = fminf(d * invn[q], 1.0f);
        termP[k] = s * INV_TEMP;
    }
}

// ---------------------------------------------------------------------------
// 6) deterministic final reduction (single block, fixed order)
// ---------------------------------------------------------------------------
__launch_bounds__(256)
__global__ void final_kernel(const float* __restrict__ termA,
                             const float* __restrict__ termP,
                             float* __restrict__ out) {
    __shared__ float red[256];
    const int t = threadIdx.x;
    float s = 0.f;
    for (int i = t; i < TWOB;  i += 256) s += termA[i];
    for (int i = t; i < NPAIR; i += 256) s -= termP[i];
    red[t] = s;
    __syncthreads();
    for (int k = 128; k > 0; k >>= 1) {
        if (t < k) red[t] += red[t + k];
        __syncthreads();
    }
    if (t == 0) out[0] = red[0] * (1.0f / 65536.0f);      // 1/(4*B*NEARBY)
}

// ---------------------------------------------------------------------------
extern "C" void kernel_launch(void* const* d_in, const int* in_sizes, int n_in,
                              void* d_out, int out_size, void* d_ws, size_t ws_size,
                              hipStream_t stream) {
    const float* z1 = (const float*)d_in[0];   // [4096,128]
    const float* z2 = (const float*)d_in[1];   // [4096,128]
    const float* zn = (const float*)d_in[2];   // [16384,128]
    float* out = (float*)d_out;

    float* zan    = (float*)d_ws;              // 8192*128 = 1048576 f
    float* rowsum = zan + (size_t)TWOB * DIM;  // 8192 f
    float* invn   = rowsum + TWOB;             // 16384 f
    float* termA  = invn + NNB;                // 8192 f
    float* termP  = termA + TWOB;              // 32768 f   (~4.25 MB total)

    norm_za_kernel    <<<TWOB / 8, 256, 0, stream>>>(z1, z2, zan);
    zn_invnorm_kernel <<<NNB / 8, 256, 0, stream>>>(zn, invn);
    gram_rowsum_kernel<<<NTILES, 256, 0, stream>>>(zan, rowsum);
    terma_kernel      <<<TWOB / 8, 256, 0, stream>>>(zan, rowsum, termA);
    termp_kernel      <<<NPAIR / 8, 256, 0, stream>>>(zan, zn, invn, termP);
    final_kernel      <<<1, 256, 0, stream>>>(termA, termP, out);
}